// SSDLoss_33328946217057
// MI455X (gfx1250) — compile-verified
//
#include <hip/hip_runtime.h>
#include <cstdint>
#include <cstddef>

#define N_ 64
#define P_ 24564
#define C_ 81
#define O_ 12
#define VAR0 0.1f
#define VAR1 0.2f
#define OVERLAP 0.5f
#define NEGPOS 3
#define TILE3 128

typedef float v8f __attribute__((ext_vector_type(8)));
typedef float v2f __attribute__((ext_vector_type(2)));
typedef float fv4 __attribute__((ext_vector_type(4)));

#if defined(__AMDGCN__) && __has_builtin(__builtin_amdgcn_global_load_async_to_lds_b128) && \
    __has_builtin(__builtin_amdgcn_s_wait_asynccnt)
#define USE_ASYNC_LDS 1
#endif

// ---------------- K0: zero accumulators ----------------
__global__ void k0_init(float* __restrict__ acc, int* __restrict__ npos) {
  int i = blockIdx.x * blockDim.x + threadIdx.x;
  if (i < 4) acc[i] = 0.f;
  if (i < N_) npos[i] = 0;
}

// ---------------- K1: best prior per (n, box) ----------------
__global__ void k1_best_prior(const float* __restrict__ prior,
                              const float* __restrict__ target,
                              int* __restrict__ bpi) {
  int b = blockIdx.x;            // n*O + j
  int n = b / O_, j = b % O_;
  const float* t = target + ((size_t)n * O_ + j) * 5;
  float bx0 = t[0], by0 = t[1], bx1 = t[2], by1 = t[3];
  float aa = (bx1 - bx0) * (by1 - by0);
  float best = -1.f; int bidx = 0;
  for (int p = threadIdx.x; p < P_; p += blockDim.x) {
    float cx = prior[p*4+0], cy = prior[p*4+1], w = prior[p*4+2], h = prior[p*4+3];
    float px0 = cx - 0.5f*w, py0 = cy - 0.5f*h, px1 = cx + 0.5f*w, py1 = cy + 0.5f*h;
    float ix = fmaxf(fminf(bx1, px1) - fmaxf(bx0, px0), 0.f);
    float iy = fmaxf(fminf(by1, py1) - fmaxf(by0, py0), 0.f);
    float inter = ix * iy;
    float iou = inter / (aa + w*h - inter);
    if (iou > best) { best = iou; bidx = p; }   // ascending p -> first-index ties
  }
  __shared__ float sv[256]; __shared__ int si[256];
  sv[threadIdx.x] = best; si[threadIdx.x] = bidx;
  __syncthreads();
  for (int s = 128; s > 0; s >>= 1) {
    if (threadIdx.x < s) {
      float v2 = sv[threadIdx.x + s]; int i2 = si[threadIdx.x + s];
      if (v2 > sv[threadIdx.x] || (v2 == sv[threadIdx.x] && i2 < si[threadIdx.x])) {
        sv[threadIdx.x] = v2; si[threadIdx.x] = i2;
      }
    }
    __syncthreads();
  }
  if (threadIdx.x == 0) bpi[b] = si[0];
}

// ---------------- K2: match + encode + smooth-L1 + pos count ----------------
__global__ void k2_match(const float* __restrict__ prior,
                         const float* __restrict__ target,
                         const float* __restrict__ loc_pred,
                         const int* __restrict__ bpi,
                         uint8_t* __restrict__ conf_ti,
                         int* __restrict__ npos,
                         float* __restrict__ acc) {
  int n = blockIdx.y;
  int p = blockIdx.x * blockDim.x + threadIdx.x;
  __shared__ float tg[O_ * 5];
  __shared__ int bp[O_];
  if (threadIdx.x < O_ * 5) tg[threadIdx.x] = target[((size_t)n * O_) * 5 + threadIdx.x];
  if (threadIdx.x < O_) bp[threadIdx.x] = bpi[n * O_ + threadIdx.x];
  __syncthreads();
  float part = 0.f; int cnt = 0;
  if (p < P_) {
    float cx = prior[p*4+0], cy = prior[p*4+1], w = prior[p*4+2], hh = prior[p*4+3];
    float px0 = cx - 0.5f*w, py0 = cy - 0.5f*hh, px1 = cx + 0.5f*w, py1 = cy + 0.5f*hh;
    float ab = w * hh;
    float best = -1.f; int bi2 = 0;
#pragma unroll
    for (int j = 0; j < O_; ++j) {
      float bx0 = tg[j*5], by0 = tg[j*5+1], bx1 = tg[j*5+2], by1 = tg[j*5+3];
      float aa = (bx1 - bx0) * (by1 - by0);
      float ix = fmaxf(fminf(bx1, px1) - fmaxf(bx0, px0), 0.f);
      float iy = fmaxf(fminf(by1, py1) - fmaxf(by0, py0), 0.f);
      float inter = ix * iy;
      float iou = inter / (aa + ab - inter);
      if (iou > best) { best = iou; bi2 = j; }
    }
#pragma unroll
    for (int j = 0; j < O_; ++j)
      if (bp[j] == p) { best = 2.f; bi2 = j; }   // scatter override, last wins
    float lab = tg[bi2*5+4];
    float conf = (best < OVERLAP) ? 0.f : (lab + 1.f);
    int t = (int)(conf + 0.5f);
    conf_ti[(size_t)n * P_ + p] = (uint8_t)t;
    if (t > 0) {
      float bx0 = tg[bi2*5], by0 = tg[bi2*5+1], bx1 = tg[bi2*5+2], by1 = tg[bi2*5+3];
      float g[4];
      g[0] = ((bx0 + bx1) * 0.5f - cx) / (VAR0 * w);
      g[1] = ((by0 + by1) * 0.5f - cy) / (VAR0 * hh);
      g[2] = __logf((bx1 - bx0) / w) / VAR1;
      g[3] = __logf((by1 - by0) / hh) / VAR1;
      const float* lp = loc_pred + ((size_t)n * P_ + p) * 4;
#pragma unroll
      for (int q = 0; q < 4; ++q) {
        float d = lp[q] - g[q];
        float ad = fabsf(d);
        part += (ad < 1.f) ? 0.5f * d * d : (ad - 0.5f);
      }
      cnt = 1;
    }
  }
  __shared__ float rf[256]; __shared__ int ri[256];
  rf[threadIdx.x] = part; ri[threadIdx.x] = cnt;
  __syncthreads();
  for (int s = 128; s > 0; s >>= 1) {
    if (threadIdx.x < s) { rf[threadIdx.x] += rf[threadIdx.x + s]; ri[threadIdx.x] += ri[threadIdx.x + s]; }
    __syncthreads();
  }
  if (threadIdx.x == 0) { atomicAdd(&acc[0], rf[0]); atomicAdd(&npos[n], ri[0]); }
}

// ---------------- K3: fused logsumexp via WMMA f32 16x16x4 ----------------
// One 509MB streaming pass over conf_pred. Tiles staged to LDS via async
// global->LDS (ASYNCcnt) when the toolchain exposes it, else nontemporal
// b128 loads. Exp row-sums accumulate through V_WMMA_F32_16X16X4_F32.
__global__ void k3_logp(const float* __restrict__ conf_pred,
                        const uint8_t* __restrict__ conf_ti,
                        float* __restrict__ loss_c,
                        float* __restrict__ acc) {
  int n = blockIdx.y;
  int p0 = blockIdx.x * TILE3;
  int rem = P_ - p0; if (rem > TILE3) rem = TILE3;

  __shared__ __align__(16) float xs[TILE3 * C_];
  __shared__ float sums[TILE3];
  __shared__ float red[256];

  const float* gbase = conf_pred + ((size_t)n * P_ + p0) * C_;
  int n4 = (rem * C_) >> 2;   // rem*81 always divisible by 4 here

  // warm GL2 for the next tile region
  if (p0 + TILE3 < P_)
    __builtin_prefetch((const char*)gbase + (size_t)TILE3 * C_ * 4 + threadIdx.x * 128, 0, 0);

#ifdef USE_ASYNC_LDS
  {
    typedef int v4i_ __attribute__((vector_size(16)));
    typedef __attribute__((address_space(3))) v4i_ LV4;
    v4i_* g = (v4i_*)gbase;          // builtin expects generic ptr to 16B vector
    LV4* l = (LV4*)xs;               // AS(3) matches exactly or converts to generic
    for (int i = threadIdx.x; i < n4; i += 256)
      __builtin_amdgcn_global_load_async_to_lds_b128(g + i, l + i, 0, 0);
    __builtin_amdgcn_s_wait_asynccnt(0);
  }
#else
  {
    const fv4* src = reinterpret_cast<const fv4*>(gbase);
    fv4* dst = reinterpret_cast<fv4*>(xs);
    for (int i = threadIdx.x; i < n4; i += 256)
      dst[i] = __builtin_nontemporal_load(src + i);   // single-use stream: TH=NT
  }
#endif
  __syncthreads();

  int lane = threadIdx.x & 31;
  int wv = threadIdx.x >> 5;
  int ql = lane & 15, h = lane >> 4;
  int qi = (wv << 4) + ql;                  // prior within tile for this lane

  // row max: half-split over classes, combined across the wave halves
  float m = -3.4e38f;
  int c0 = h ? 41 : 0, c1 = h ? C_ : 41;
  for (int c = c0; c < c1; ++c) m = fmaxf(m, xs[qi * C_ + c]);
  m = fmaxf(m, __shfl_xor(m, 16, 32));

  // WMMA chain: A lane layout (ISA 7.12.2): lanes 0-15 -> K=0,1 ; lanes 16-31 -> K=2,3
  v8f accv = {0.f, 0.f, 0.f, 0.f, 0.f, 0.f, 0.f, 0.f};
  v2f bb = {1.f, 1.f};
#pragma unroll
  for (int j = 0; j < 21; ++j) {
    int k0 = 4 * j + 2 * h;
    v2f a;
    a.x = (k0 < C_)     ? __expf(xs[qi * C_ + k0]     - m) : 0.f;
    a.y = (k0 + 1 < C_) ? __expf(xs[qi * C_ + k0 + 1] - m) : 0.f;
    accv = __builtin_amdgcn_wmma_f32_16x16x4_f32(false, a, false, bb,
                                                 (short)0, accv, false, false);
  }
  // B==ones => every column of D holds the row sum; lane ql==0 of each half
  // carries M=0..7 (h=0) / M=8..15 (h=1) in accv[0..7]
  if (ql == 0) {
#pragma unroll
    for (int r = 0; r < 8; ++r) sums[(wv << 4) + (h << 3) + r] = accv[r];
  }
  __syncthreads();

  float part = 0.f;
  int p = p0 + qi;
  if (h == 0 && qi < rem) {
    float s = sums[qi];
    int t = conf_ti[(size_t)n * P_ + p];
    float lp = m + __logf(s) - xs[qi * C_ + t];   // logsumexp - x[target] == ce == logp
    bool pos = t > 0;
    loss_c[(size_t)n * P_ + p] = pos ? 0.f : lp;
    if (pos) part = lp;
  }
  red[threadIdx.x] = part;
  __syncthreads();
  for (int s2 = 128; s2 > 0; s2 >>= 1) {
    if (threadIdx.x < s2) red[threadIdx.x] += red[threadIdx.x + s2];
    __syncthreads();
  }
  if (threadIdx.x == 0) atomicAdd(&acc[1], red[0]);
}

// ---------------- K4: per-row radix select (k-th largest of loss_c) ----------------
__global__ void k4_select(const float* __restrict__ loss_c,
                          const int* __restrict__ npos,
                          float* __restrict__ Trow,
                          int* __restrict__ needrow) {
  int n = blockIdx.x;
  __shared__ int hist[256];
  __shared__ unsigned sPrefix, sMask; __shared__ int sRemain;
  int k = npos[n] * NEGPOS; if (k > P_ - 1) k = P_ - 1;
  if (k <= 0) {
    if (threadIdx.x == 0) { Trow[n] = __int_as_float(0x7f800000); needrow[n] = 0; }
    return;
  }
  unsigned prefix = 0, mask = 0; int remain = k;
  for (int lvl = 3; lvl >= 0; --lvl) {
    hist[threadIdx.x] = 0;
    __syncthreads();
    for (int p = threadIdx.x; p < P_; p += 256) {
      unsigned bits = __float_as_uint(loss_c[(size_t)n * P_ + p]);  // >=0: bit-monotone
      if ((bits & mask) == prefix) atomicAdd(&hist[(bits >> (lvl * 8)) & 255], 1);
    }
    __syncthreads();
    if (threadIdx.x == 0) {
      int cum = 0, b = 255;
      for (; b > 0; --b) { cum += hist[b]; if (cum >= remain) break; }
      if (cum < remain) cum += hist[0];            // b==0 fallthrough
      sRemain = remain - (cum - hist[b]);
      sPrefix = prefix | ((unsigned)b << (lvl * 8));
      sMask = mask | (0xFFu << (lvl * 8));
    }
    __syncthreads();
    prefix = sPrefix; mask = sMask; remain = sRemain;
    __syncthreads();
  }
  if (threadIdx.x == 0) { Trow[n] = __uint_as_float(prefix); needrow[n] = remain; }
}

// ---------------- K5: sum of loss_c strictly above per-row threshold ----------------
__global__ void k5_negsum(const float* __restrict__ loss_c,
                          const float* __restrict__ Trow,
                          float* __restrict__ acc) {
  int n = blockIdx.y;
  int p = blockIdx.x * blockDim.x + threadIdx.x;
  float T = Trow[n];
  float part = 0.f;
  if (p < P_) { float v = loss_c[(size_t)n * P_ + p]; if (v > T) part = v; }
  __shared__ float red[256];
  red[threadIdx.x] = part;
  __syncthreads();
  for (int s = 128; s > 0; s >>= 1) {
    if (threadIdx.x < s) red[threadIdx.x] += red[threadIdx.x + s];
    __syncthreads();
  }
  if (threadIdx.x == 0) atomicAdd(&acc[2], red[0]);
}

// ---------------- K6: finalize the two scalars ----------------
__global__ void k6_final(const float* __restrict__ acc,
                         const int* __restrict__ npos,
                         const float* __restrict__ Trow,
                         const int* __restrict__ needrow,
                         float* __restrict__ out) {
  if (threadIdx.x == 0 && blockIdx.x == 0) {
    float tp = 0.f, extra = 0.f;
    for (int n = 0; n < N_; ++n) {
      tp += (float)npos[n];
      if (needrow[n] > 0) extra += (float)needrow[n] * Trow[n];  // ties contribute T each
    }
    out[0] = acc[0] / tp;
    out[1] = (acc[1] + acc[2] + extra) / tp;
  }
}

// ---------------- host launcher ----------------
static inline size_t alignUp(size_t x, size_t a) { return (x + a - 1) & ~(a - 1); }

extern "C" void kernel_launch(void* const* d_in, const int* in_sizes, int n_in,
                              void* d_out, int out_size, void* d_ws, size_t ws_size,
                              hipStream_t stream) {
  const float* loc_pred  = (const float*)d_in[0];
  const float* conf_pred = (const float*)d_in[1];
  const float* prior     = (const float*)d_in[2];
  const float* target    = (const float*)d_in[3];
  float* out = (float*)d_out;

  char* ws = (char*)d_ws;
  size_t off = 0;
  float* acc = (float*)(ws + off);       off = alignUp(off + 4 * sizeof(float), 16);
  int* npos  = (int*)(ws + off);         off = alignUp(off + N_ * sizeof(int), 16);
  int* bpi   = (int*)(ws + off);         off = alignUp(off + N_ * O_ * sizeof(int), 16);
  float* Trow = (float*)(ws + off);      off = alignUp(off + N_ * sizeof(float), 16);
  int* needrow = (int*)(ws + off);       off = alignUp(off + N_ * sizeof(int), 16);
  uint8_t* cti = (uint8_t*)(ws + off);   off = alignUp(off + (size_t)N_ * P_, 16);
  float* loss_c = (float*)(ws + off);    off = alignUp(off + (size_t)N_ * P_ * sizeof(float), 16);

  k0_init<<<1, 256, 0, stream>>>(acc, npos);
  k1_best_prior<<<N_ * O_, 256, 0, stream>>>(prior, target, bpi);
  dim3 g2((P_ + 255) / 256, N_);
  k2_match<<<g2, 256, 0, stream>>>(prior, target, loc_pred, bpi, cti, npos, acc);
  dim3 g3((P_ + TILE3 - 1) / TILE3, N_);
  k3_logp<<<g3, 256, 0, stream>>>(conf_pred, cti, loss_c, acc);
  k4_select<<<N_, 256, 0, stream>>>(loss_c, npos, Trow, needrow);
  k5_negsum<<<g2, 256, 0, stream>>>(loss_c, Trow, acc);
  k6_final<<<1, 1, 0, stream>>>(acc, npos, Trow, needrow, out);
}